// E_GCL_AT_X_11751030522788
// MI455X (gfx1250) — compile-verified
//
#include <hip/hip_runtime.h>
#include <hip/hip_bf16.h>

// N=768, L=8, H=64, K=4, T=3
#define NN   768
#define LL   8
#define HH   64
#define NL   6144          // N*L rows
#define QKSZ 393216        // L*N*H floats
#define GTW  16            // packed G columns: 4 (S1) + 12 (S2)

typedef __attribute__((ext_vector_type(2))) float v2f;
typedef __attribute__((ext_vector_type(8))) float v8f;

__device__ __forceinline__ v8f wmma4(v2f a, v2f b, v8f c) {
    // V_WMMA_F32_16X16X4_F32 : D = A(16x4) * B(4x16) + C(16x16)
    return __builtin_amdgcn_wmma_f32_16x16x4_f32(
        /*neg_a=*/false, a, /*neg_b=*/false, b,
        /*c_mod=*/(short)0, c, /*reuse_a=*/false, /*reuse_b=*/false);
}

// ---------------------------------------------------------------------------
// Kernel A: q/k/v MLPs + cmv, writes frame-major layouts for the attn kernel.
//   q,k  : [L][N][H]    (row-major per frame)
//   vT   : [L][H][N]    (so alpha@v B-operand is contiguous in K=j)
//   GT   : [L][16][N]   (packed S1|S2 B-operand, contiguous in K=j)
// ---------------------------------------------------------------------------
__global__ __launch_bounds__(128)
void qkv_kernel(const float* __restrict__ hin, const float* __restrict__ coord,
                const float* __restrict__ wq1, const float* __restrict__ bq1,
                const float* __restrict__ wq2, const float* __restrict__ bq2,
                const float* __restrict__ wk1, const float* __restrict__ bk1,
                const float* __restrict__ wk2, const float* __restrict__ bk2,
                const float* __restrict__ wv1, const float* __restrict__ bv1,
                const float* __restrict__ wv2, const float* __restrict__ bv2,
                const float* __restrict__ wc1, const float* __restrict__ bc1,
                const float* __restrict__ wc2,
                float* __restrict__ qo, float* __restrict__ ko,
                float* __restrict__ vTo, float* __restrict__ GT)
{
    __shared__ float XS[16][68];   // block's 16 input rows (A of layer1)
    __shared__ float TS[16][68];   // layer-1 output (A of layer2)
    __shared__ float VS[16][68];   // v result (A of coord-mlp layer1)
    __shared__ float WS[64 * 64];  // current weight matrix
    __shared__ float CMV[16][4];   // cmv tile

    const int tid   = threadIdx.x;
    const int lane  = tid & 31;
    const int wave  = tid >> 5;        // 0..3, owns cols n0 = wave*16
    const int r0    = blockIdx.x * 16; // first flattened (i,l) row
    const int m16   = lane & 15;
    const int khalf = (lane >> 4) * 2;
    const int mb    = (lane >> 4) * 8;
    const int n0    = wave * 16;

    // stage the 16 input rows
    {
        int m = tid >> 3, c0 = (tid & 7) * 8;
        #pragma unroll
        for (int c = 0; c < 8; ++c)
            XS[m][c0 + c] = hin[(r0 + m) * HH + c0 + c];
    }

    const float* W1s[3] = {wq1, wk1, wv1};
    const float* B1s[3] = {bq1, bk1, bv1};
    const float* W2s[3] = {wq2, wk2, wv2};
    const float* B2s[3] = {bq2, bk2, bv2};

    for (int hh = 0; hh < 3; ++hh) {
        __syncthreads();
        #pragma unroll
        for (int i = 0; i < 32; ++i) WS[i * 128 + tid] = W1s[hh][i * 128 + tid];
        __syncthreads();

        // layer 1: relu(X @ W1 + b1) -> TS
        v8f acc = {};
        #pragma unroll
        for (int kk = 0; kk < 64; kk += 4) {
            v2f a; a.x = XS[m16][kk + khalf]; a.y = XS[m16][kk + khalf + 1];
            v2f b; b.x = WS[(kk + khalf) * 64 + n0 + m16];
                   b.y = WS[(kk + khalf + 1) * 64 + n0 + m16];
            acc = wmma4(a, b, acc);
        }
        float bb = B1s[hh][n0 + m16];
        #pragma unroll
        for (int r = 0; r < 8; ++r) {
            float v = acc[r] + bb;
            TS[mb + r][n0 + m16] = v > 0.f ? v : 0.f;
        }
        __syncthreads();
        #pragma unroll
        for (int i = 0; i < 32; ++i) WS[i * 128 + tid] = W2s[hh][i * 128 + tid];
        __syncthreads();

        // layer 2: relu(T @ W2 + b2)
        v8f acc2 = {};
        #pragma unroll
        for (int kk = 0; kk < 64; kk += 4) {
            v2f a; a.x = TS[m16][kk + khalf]; a.y = TS[m16][kk + khalf + 1];
            v2f b; b.x = WS[(kk + khalf) * 64 + n0 + m16];
                   b.y = WS[(kk + khalf + 1) * 64 + n0 + m16];
            acc2 = wmma4(a, b, acc2);
        }
        float bb2 = B2s[hh][n0 + m16];
        #pragma unroll
        for (int r = 0; r < 8; ++r) {
            float v = acc2[r] + bb2;
            v = v > 0.f ? v : 0.f;
            int rowg = r0 + mb + r;
            int i = rowg >> 3, l = rowg & 7;
            if (hh == 0)      qo[l * (NN * HH) + i * HH + n0 + m16] = v;
            else if (hh == 1) ko[l * (NN * HH) + i * HH + n0 + m16] = v;
            else {
                vTo[l * (HH * NN) + (n0 + m16) * NN + i] = v;
                VS[mb + r][n0 + m16] = v;
            }
        }
    }

    // coord_mlp(v): layer 1 relu(V @ wc1 + bc1) -> TS
    __syncthreads();
    #pragma unroll
    for (int i = 0; i < 32; ++i) WS[i * 128 + tid] = wc1[i * 128 + tid];
    __syncthreads();
    {
        v8f acc = {};
        #pragma unroll
        for (int kk = 0; kk < 64; kk += 4) {
            v2f a; a.x = VS[m16][kk + khalf]; a.y = VS[m16][kk + khalf + 1];
            v2f b; b.x = WS[(kk + khalf) * 64 + n0 + m16];
                   b.y = WS[(kk + khalf + 1) * 64 + n0 + m16];
            acc = wmma4(a, b, acc);
        }
        float bb = bc1[n0 + m16];
        #pragma unroll
        for (int r = 0; r < 8; ++r) {
            float v = acc[r] + bb;
            TS[mb + r][n0 + m16] = v > 0.f ? v : 0.f;
        }
    }
    __syncthreads();

    // layer 2: cmv = T @ wc2  (only 4 valid cols; wave 0 computes, B padded w/ 0)
    if (wave == 0) {
        v8f acc = {};
        #pragma unroll
        for (int kk = 0; kk < 64; kk += 4) {
            v2f a; a.x = TS[m16][kk + khalf]; a.y = TS[m16][kk + khalf + 1];
            v2f b;
            b.x = (m16 < 4) ? wc2[(kk + khalf) * 4 + m16] : 0.f;
            b.y = (m16 < 4) ? wc2[(kk + khalf + 1) * 4 + m16] : 0.f;
            acc = wmma4(a, b, acc);
        }
        if (m16 < 4) {
            #pragma unroll
            for (int r = 0; r < 8; ++r) CMV[mb + r][m16] = acc[r];
        }
    }
    __syncthreads();

    // Build GT[l][c][i]: c<4 -> cmv ; c>=4 -> cmv[k]*coord[k][t], kt = c-4
    for (int e = tid; e < 256; e += 128) {
        int m = e >> 4, c = e & 15;
        int rowg = r0 + m, i = rowg >> 3, l = rowg & 7;
        float val;
        if (c < 4) val = CMV[m][c];
        else {
            int kt = c - 4, kk = kt / 3, t = kt - kk * 3;
            val = CMV[m][kk] * coord[rowg * 12 + kk * 3 + t];
        }
        GT[l * (GTW * NN) + c * NN + i] = val;
    }
}

// ---------------------------------------------------------------------------
// Kernel B: per (frame l, 16-row i-block): scores -> softmax -> alpha@v and
// alpha@G, fused epilogue for both outputs. 8 waves / 256 threads.
// ---------------------------------------------------------------------------
__global__ __launch_bounds__(256)
void attn_kernel(const float* __restrict__ hin, const float* __restrict__ coord,
                 const float* __restrict__ q, const float* __restrict__ k,
                 const float* __restrict__ vT, const float* __restrict__ GT,
                 float* __restrict__ out_h, float* __restrict__ out_x)
{
    __shared__ float SA[16 * 772];    // exp(scores) stripe, padded stride
    __shared__ float red[16][16];
    __shared__ float rmaxs[16];
    __shared__ float rinv[16];
    __shared__ float GS[16][16];

    const int tid   = threadIdx.x;
    const int lane  = tid & 31;
    const int wave  = tid >> 5;                 // 0..7
    const int l     = blockIdx.x & 7;
    const int i0    = (blockIdx.x >> 3) * 16;
    const int m16   = lane & 15;
    const int khalf = (lane >> 4) * 2;
    const int mb    = (lane >> 4) * 8;

    const float* ql = q  + l * (NN * HH);
    const float* kl = k  + l * (NN * HH);
    const float* vl = vT + l * (HH * NN);
    const float* Gl = GT + l * (GTW * NN);

    // ---- phase 1: scores[i0:i0+16, :] ;  q.kT => both operands contiguous
    const float* arow = ql + (i0 + m16) * HH;
    for (int u = 0; u < 6; ++u) {
        int j0 = (wave * 6 + u) * 16;
        const float* brow = kl + (j0 + m16) * HH;
        v8f acc = {};
        #pragma unroll
        for (int kk = 0; kk < 64; kk += 4) {
            v2f a = *(const v2f*)(arow + kk + khalf);
            v2f b = *(const v2f*)(brow + kk + khalf);
            acc = wmma4(a, b, acc);
        }
        #pragma unroll
        for (int r = 0; r < 8; ++r)
            SA[(mb + r) * 772 + j0 + m16] = acc[r];
    }
    __syncthreads();

    // ---- phase 2: softmax over j (row-wise); keep unnormalized exp, fold
    //      1/sum into the output scaling.
    {
        int row = tid >> 4, c0 = tid & 15;
        float pm = -3.4e38f;
        for (int s = 0; s < 48; ++s)
            pm = fmaxf(pm, SA[row * 772 + c0 + 16 * s]);
        red[row][c0] = pm;
        __syncthreads();
        if (tid < 16) {
            float mv = red[tid][0];
            #pragma unroll
            for (int j = 1; j < 16; ++j) mv = fmaxf(mv, red[tid][j]);
            rmaxs[tid] = mv;
        }
        __syncthreads();
        float rm = rmaxs[row], ps = 0.f;
        for (int s = 0; s < 48; ++s) {
            float e = __expf(SA[row * 772 + c0 + 16 * s] - rm);
            SA[row * 772 + c0 + 16 * s] = e;
            ps += e;
        }
        red[row][c0] = ps;
        __syncthreads();
        if (tid < 16) {
            float sum = 0.f;
            #pragma unroll
            for (int j = 0; j < 16; ++j) sum += red[tid][j];
            rinv[tid] = 1.0f / sum;
        }
        __syncthreads();
    }

    // ---- phase 3: waves 0-3 -> h_agg cols; wave 4 -> packed S1|S2 tile
    if (wave < 4) {
        int n0 = wave * 16;
        const float* bcol = vl + (n0 + m16) * NN;
        v8f acc = {};
        #pragma unroll 4
        for (int kk = 0; kk < NN; kk += 4) {
            v2f a = *(const v2f*)(&SA[m16 * 772 + kk + khalf]);
            v2f b = *(const v2f*)(bcol + kk + khalf);
            acc = wmma4(a, b, acc);
        }
        #pragma unroll
        for (int r = 0; r < 8; ++r) {
            int m = mb + r;
            int gi = ((i0 + m) * LL + l) * HH + n0 + m16;
            out_h[gi] = hin[gi] + acc[r] * rinv[m];
        }
    } else if (wave == 4) {
        const float* bcol = Gl + m16 * NN;
        v8f acc = {};
        #pragma unroll 4
        for (int kk = 0; kk < NN; kk += 4) {
            v2f a = *(const v2f*)(&SA[m16 * 772 + kk + khalf]);
            v2f b = *(const v2f*)(bcol + kk + khalf);
            acc = wmma4(a, b, acc);
        }
        #pragma unroll
        for (int r = 0; r < 8; ++r)
            GS[mb + r][m16] = acc[r] * rinv[mb + r];
    }
    __syncthreads();

    // ---- coord epilogue: out_x = coord*(1+S1) - S2
    for (int e = tid; e < 192; e += 256) {
        int m = e / 12, kt = e - m * 12;
        int kk = kt / 3;
        float S1 = GS[m][kk];
        float S2 = GS[m][4 + kt];
        int gi = ((i0 + m) * LL + l) * 12 + kt;
        out_x[gi] = coord[gi] * (1.f + S1) - S2;
    }
}

extern "C" void kernel_launch(void* const* d_in, const int* in_sizes, int n_in,
                              void* d_out, int out_size, void* d_ws, size_t ws_size,
                              hipStream_t stream) {
    const float* hin   = (const float*)d_in[0];
    const float* coord = (const float*)d_in[1];
    const float* wq1 = (const float*)d_in[2];  const float* bq1 = (const float*)d_in[3];
    const float* wq2 = (const float*)d_in[4];  const float* bq2 = (const float*)d_in[5];
    const float* wk1 = (const float*)d_in[6];  const float* bk1 = (const float*)d_in[7];
    const float* wk2 = (const float*)d_in[8];  const float* bk2 = (const float*)d_in[9];
    const float* wv1 = (const float*)d_in[10]; const float* bv1 = (const float*)d_in[11];
    const float* wv2 = (const float*)d_in[12]; const float* bv2 = (const float*)d_in[13];
    const float* wc1 = (const float*)d_in[14]; const float* bc1 = (const float*)d_in[15];
    const float* wc2 = (const float*)d_in[16];

    float* ws = (float*)d_ws;
    float* qbuf = ws;                 // [L][N][H]
    float* kbuf = ws + QKSZ;          // [L][N][H]
    float* vTb  = ws + 2 * QKSZ;      // [L][H][N]
    float* GTb  = ws + 3 * QKSZ;      // [L][16][N]

    float* out_h = (float*)d_out;          // [N][L][H]
    float* out_x = out_h + QKSZ;           // [N][L][K][T]

    qkv_kernel<<<NL / 16, 128, 0, stream>>>(hin, coord,
        wq1, bq1, wq2, bq2, wk1, bk1, wk2, bk2, wv1, bv1, wv2, bv2,
        wc1, bc1, wc2, qbuf, kbuf, vTb, GTb);

    attn_kernel<<<LL * (NN / 16), 256, 0, stream>>>(hin, coord,
        qbuf, kbuf, vTb, GTb, out_h, out_x);
}